// ProjectLayerWithMask_84499186581698
// MI455X (gfx1250) — compile-verified
//
#include <hip/hip_runtime.h>

typedef float v2f __attribute__((ext_vector_type(2)));
typedef float v8f __attribute__((ext_vector_type(8)));

constexpr int KCAM = 5, BB = 4, CC = 15, HH = 128, WW = 240, CUBE = 64;
constexpr int NVOX = CUBE * CUBE * CUBE;          // 262144
constexpr int HWSZ = HH * WW;                     // 30720
constexpr int BLOCK = 256;
constexpr int BLOCKS_PER_B = NVOX / BLOCK;        // 1024
constexpr float INVK = 1.0f / (5.0f + 1e-6f);
// fold: g = clip(xy2 * (HM/IMG) / (HM-1) * 2 - 1, -1.1, 1.1); HM/IMG = 0.25 both axes
constexpr float GXS = 0.5f / 239.0f;
constexpr float GYS = 0.5f / 127.0f;

struct SCam {
    float P[3][4];                 // [R | -R*T] homogeneous projection rows
    float fx, fy, cx, cy, maxwh;
    float t[6];                    // 2x3 affine
};
struct SBatch {
    SCam  cam[KCAM];
    float wb2[3], wsc[3];          // world = wb2 + idx*wsc (centers_tl folded in)
    int   lo[3], hi[3];            // valid window relative to centers_tl
    float validf;
};

__global__ __launch_bounds__(BLOCK)
void ProjectLayerWithMask_kernel(
    const float* __restrict__ fm,    // [K,B,C,H,W]
    const float* __restrict__ gc,    // [B,7]
    const float* __restrict__ sc,    // [3]
    const float* __restrict__ ss,    // [3]
    const float* __restrict__ sub,   // [3]
    const float* __restrict__ cent,  // [B,K,2]
    const float* __restrict__ trans, // [B,K,2,3]
    const float* __restrict__ Rm,    // [B,K,3,3]
    const float* __restrict__ Tm,    // [B,K,3]
    const float* __restrict__ fv,    // [B,K,2]
    const float* __restrict__ cv,    // [B,K,2]
    float* __restrict__ outCubes,    // [B,C,64^3]
    float* __restrict__ outOff)      // [B,3]
{
    __shared__ SBatch s;
    const int b   = blockIdx.x / BLOCKS_PER_B;
    const int blk = blockIdx.x % BLOCKS_PER_B;
    const int tid = threadIdx.x;

    // ---- per-batch scalar setup (once per block, tiny) ----
    if (tid == 0) {
        #pragma unroll
        for (int d = 0; d < 3; ++d) {
            float ssd = ss[d], subd = sub[d], scd = sc[d];
            int   fine = (int)(ssd / subd * (float)(CUBE - 1)) + 1;   // astype(int32) trunc
            float fnm1 = (float)(fine - 1);
            float scale = fnm1 / ssd;
            float bias  = -subd * 0.5f / ssd * fnm1 - scale * (scd - ssd * 0.5f);
            int   ctl   = (int)rintf(gc[b * 7 + d] * scale + bias);   // round-half-even
            if (blk == 0)
                outOff[b * 3 + d] = (float)ctl / fnm1 * ssd - ssd * 0.5f + subd * 0.5f;
            int mask = 0;
            if (d < 2) {
                int mm = (int)((1.0f - gc[b * 7 + 5 + d]) * 0.5f * (float)(CUBE - 1));
                mask = mm > 0 ? mm : 0;
            }
            int st = ctl + mask;        if (st < 0) st = 0;
            int en = ctl + CUBE - mask; if (en > fine) en = fine;
            if (en < st + 1) en = st + 1;
            s.lo[d]  = st - ctl;
            s.hi[d]  = en - ctl;
            s.wsc[d] = ssd / fnm1;
            s.wb2[d] = scd - ssd * 0.5f + (float)ctl * (ssd / fnm1);
        }
        s.validf = (gc[b * 7 + 3] >= 0.0f) ? 1.0f : 0.0f;
    }
    if (tid >= 1 && tid <= KCAM) {
        int q = tid - 1;
        int base = b * KCAM + q;
        const float* R = Rm + base * 9;
        const float* T = Tm + base * 3;
        #pragma unroll
        for (int r = 0; r < 3; ++r) {
            float r0 = R[r * 3 + 0], r1 = R[r * 3 + 1], r2 = R[r * 3 + 2];
            s.cam[q].P[r][0] = r0; s.cam[q].P[r][1] = r1; s.cam[q].P[r][2] = r2;
            s.cam[q].P[r][3] = -(r0 * T[0] + r1 * T[1] + r2 * T[2]);
        }
        s.cam[q].fx = fv[base * 2]; s.cam[q].fy = fv[base * 2 + 1];
        s.cam[q].cx = cv[base * 2]; s.cam[q].cy = cv[base * 2 + 1];
        s.cam[q].maxwh = 2.0f * fmaxf(cent[base * 2], cent[base * 2 + 1]);
        #pragma unroll
        for (int i2 = 0; i2 < 6; ++i2) s.cam[q].t[i2] = trans[base * 6 + i2];
    }
    __syncthreads();

    // ---- voxel coordinates: n fastest dim = dim2(z) -> contiguous output stores ----
    const int n   = blk * BLOCK + tid;
    const int kz  = n & (CUBE - 1);
    const int jy  = (n >> 6) & (CUBE - 1);
    const int ix3 = n >> 12;

    const float validm =
        (ix3 >= s.lo[0] && ix3 < s.hi[0] && jy >= s.lo[1] && jy < s.hi[1] &&
         kz >= s.lo[2] && kz < s.hi[2]) ? s.validf : 0.0f;

    const float wxv = s.wb2[0] + (float)ix3 * s.wsc[0];
    const float wyv = s.wb2[1] + (float)jy  * s.wsc[1];
    const float wzv = s.wb2[2] + (float)kz  * s.wsc[2];

    // ---- WMMA 16x16x4 f32: D = P(15x4, 5 cams stacked) x [world;1](4x16) ----
    // A layout (16x4): lanes0-15 M rows, K0/K1 in vgpr0/1; lanes16-31 K2/K3.
    const int lane = tid & 31;
    const int m    = lane & 15;
    float a0 = 0.0f, a1 = 0.0f;
    if (m < 15) {
        int q = m / 3, r = m - q * 3;
        int k0 = (lane < 16) ? 0 : 2;
        a0 = s.cam[q].P[r][k0];
        a1 = s.cam[q].P[r][k0 + 1];
    }
    v2f A; A[0] = a0; A[1] = a1;
    v8f Z8 = {0.f, 0.f, 0.f, 0.f, 0.f, 0.f, 0.f, 0.f};

    // B layout (4x16, mirrored): vgpr0 = {K0 | K2}, vgpr1 = {K1 | K3} per lane-half
    v8f D[2];
    #pragma unroll
    for (int g = 0; g < 2; ++g) {
        int src = g * 16 + m;
        float bx = __shfl(wxv, src, 32);
        float by = __shfl(wyv, src, 32);
        float bz = __shfl(wzv, src, 32);
        v2f Bv;
        Bv[0] = (lane < 16) ? bx : bz;
        Bv[1] = (lane < 16) ? by : 1.0f;
        D[g] = __builtin_amdgcn_wmma_f32_16x16x4_f32(false, A, false, Bv,
                                                     (short)0, Z8, false, false);
    }
    // D element (m,n): vgpr m%8, lane n + 16*(m>=8). Gather my point's 15 rows.
    float Xc[15];
    #pragma unroll
    for (int r = 0; r < 15; ++r) {
        const int vi   = (r < 8) ? r : r - 8;
        const int srcl = m + ((r < 8) ? 0 : 16);
        float v0 = __shfl(D[0][vi], srcl, 32);   // valid for lanes 0-15 (points 0-15)
        float v1 = __shfl(D[1][vi], srcl, 32);   // valid for lanes 16-31
        Xc[r] = (lane < 16) ? v0 : v1;
    }

    // ---- per-camera bilinear setup: 4 clamped addresses + OOB-zeroed weights ----
    int   addr[KCAM][4];
    float wgt[KCAM][4];
    #pragma unroll
    for (int q = 0; q < KCAM; ++q) {
        float X = Xc[q * 3 + 0], Y = Xc[q * 3 + 1], Zc = Xc[q * 3 + 2];
        float inv = 1.0f / Zc;
        float px = s.cam[q].fx * X * inv + s.cam[q].cx;
        float py = s.cam[q].fy * Y * inv + s.cam[q].cy;
        float mw = s.cam[q].maxwh;
        px = fminf(fmaxf(px, -1.0f), mw);
        py = fminf(fmaxf(py, -1.0f), mw);
        float x2 = s.cam[q].t[0] * px + s.cam[q].t[1] * py + s.cam[q].t[2];
        float y2 = s.cam[q].t[3] * px + s.cam[q].t[4] * py + s.cam[q].t[5];
        float gx = fminf(fmaxf(x2 * GXS - 1.0f, -1.1f), 1.1f);
        float gy = fminf(fmaxf(y2 * GYS - 1.0f, -1.1f), 1.1f);
        float fix = (gx + 1.0f) * ((float)WW * 0.5f) - 0.5f;
        float fiy = (gy + 1.0f) * ((float)HH * 0.5f) - 0.5f;
        float x0f = floorf(fix), y0f = floorf(fiy);
        float fx1 = fix - x0f,  fy1 = fiy - y0f;
        int x0 = (int)x0f, y0 = (int)y0f, x1 = x0 + 1, y1 = y0 + 1;
        float vx0 = (x0 >= 0 && x0 < WW) ? 1.f : 0.f;
        float vx1 = (x1 >= 0 && x1 < WW) ? 1.f : 0.f;
        float vy0 = (y0 >= 0 && y0 < HH) ? 1.f : 0.f;
        float vy1 = (y1 >= 0 && y1 < HH) ? 1.f : 0.f;
        int x0c = min(max(x0, 0), WW - 1), x1c = min(max(x1, 0), WW - 1);
        int y0c = min(max(y0, 0), HH - 1), y1c = min(max(y1, 0), HH - 1);
        wgt[q][0] = (1.0f - fx1) * (1.0f - fy1) * vx0 * vy0;
        wgt[q][1] = fx1 * (1.0f - fy1) * vx1 * vy0;
        wgt[q][2] = (1.0f - fx1) * fy1 * vx0 * vy1;
        wgt[q][3] = fx1 * fy1 * vx1 * vy1;
        addr[q][0] = y0c * WW + x0c;
        addr[q][1] = y0c * WW + x1c;
        addr[q][2] = y1c * WW + x0c;
        addr[q][3] = y1c * WW + x1c;
    }

    // ---- channel loop: 5 cams x 4 L2-resident gathers, NT coalesced store ----
    const float  vm = validm * INVK;
    const size_t outBase = (size_t)b * CC * NVOX + (size_t)n;
    #pragma unroll 1
    for (int c = 0; c < CC; ++c) {
        float acc = 0.0f;
        #pragma unroll
        for (int q = 0; q < KCAM; ++q) {
            const float* p = fm + (size_t)((q * BB + b) * CC + c) * HWSZ;
            acc += wgt[q][0] * p[addr[q][0]];
            acc += wgt[q][1] * p[addr[q][1]];
            acc += wgt[q][2] * p[addr[q][2]];
            acc += wgt[q][3] * p[addr[q][3]];
        }
        float r = fminf(fmaxf(acc * vm, 0.0f), 1.0f);
        __builtin_nontemporal_store(r, outCubes + outBase + (size_t)c * NVOX);
    }
}

extern "C" void kernel_launch(void* const* d_in, const int* in_sizes, int n_in,
                              void* d_out, int out_size, void* d_ws, size_t ws_size,
                              hipStream_t stream) {
    const float* fm    = (const float*)d_in[0];   // feature_maps [5,4,15,128,240]
    const float* gc    = (const float*)d_in[1];   // grid_center [4,7]
    const float* sc    = (const float*)d_in[2];   // space_center [3]
    const float* ss    = (const float*)d_in[3];   // space_size [3]
    const float* sub   = (const float*)d_in[4];   // sub_space_size [3]
    // d_in[5] = sub_cube_size (int, ==64, compiled in)
    const float* cent  = (const float*)d_in[6];   // centers [4,5,2]
    const float* trans = (const float*)d_in[7];   // trans [4,5,2,3]
    const float* Rm    = (const float*)d_in[8];   // cam_R [4,5,3,3]
    const float* Tm    = (const float*)d_in[9];   // cam_T [4,5,3]
    const float* fv    = (const float*)d_in[10];  // cam_f [4,5,2]
    const float* cv    = (const float*)d_in[11];  // cam_c [4,5,2]

    float* outCubes = (float*)d_out;
    float* outOff   = (float*)d_out + (size_t)BB * CC * NVOX;

    dim3 grid(BB * BLOCKS_PER_B);   // 4096 blocks
    dim3 block(BLOCK);
    ProjectLayerWithMask_kernel<<<grid, block, 0, stream>>>(
        fm, gc, sc, ss, sub, cent, trans, Rm, Tm, fv, cv, outCubes, outOff);
}